// BaseQuantizer_8383776161975
// MI455X (gfx1250) — compile-verified
//
#include <hip/hip_runtime.h>
#include <hip/hip_bf16.h>

typedef __attribute__((ext_vector_type(2))) float v2f;
typedef __attribute__((ext_vector_type(4))) float v4f;
typedef __attribute__((ext_vector_type(8))) float v8f;

#define D_DIM 128
#define K_CB  1024
#define NTILES (K_CB / 16)        // 64 N-tiles of 16 codes
#define WAVES_PER_WG 4
#define THREADS_PER_WG (WAVES_PER_WG * 32)            // 128
#define TOK_PER_WAVE 32                               // two 16-row A tiles
#define TOK_PER_WG   (WAVES_PER_WG * TOK_PER_WAVE)    // 128
#define N_TOKENS (8 * 4096)       // B*L = 32768
#define TOTAL_ELEMS (N_TOKENS * D_DIM)

#define WMMA_F32(a, b, c) \
    __builtin_amdgcn_wmma_f32_16x16x4_f32(false, (a), false, (b), (short)0, (c), false, false)

// ---------------------------------------------------------------------------
// Prep: cb_sq[k] = sum_d codebook[d][k]^2 ; cbT[k][d] = codebook[d][k] ;
// zero the loss accumulator. 1024 threads (4 blocks x 256).
// ---------------------------------------------------------------------------
__global__ __launch_bounds__(256) void vq_prep_kernel(
    const float* __restrict__ cb, float* __restrict__ cbT,
    float* __restrict__ cbsq, float* __restrict__ lossSlot)
{
    int k = blockIdx.x * blockDim.x + threadIdx.x;   // 0..1023
    if (k == 0) *lossSlot = 0.0f;
    float s = 0.0f;
    #pragma unroll 4
    for (int d = 0; d < D_DIM; ++d) {
        float v = cb[(size_t)d * K_CB + k];          // lanes read contiguous k
        cbT[(size_t)k * D_DIM + d] = v;
        s = fmaf(v, v, s);
    }
    cbsq[k] = s;
}

// ---------------------------------------------------------------------------
// Main VQ kernel: 4 waves/WG, 32 tokens/wave (2 A-tiles), full K scan.
// Each LDS B pair now feeds 4 WMMAs (2x operand reuse vs 16-token waves).
// Double-buffered LDS B-tiles staged cooperatively by the whole WG.
// ---------------------------------------------------------------------------
__global__ __launch_bounds__(THREADS_PER_WG) void vq_main_kernel(
    const float* __restrict__ x,      // [32768][128]
    const float* __restrict__ cb,     // [128][1024]
    const float* __restrict__ cbT,    // [1024][128]
    const float* __restrict__ cbsq,   // [1024]
    float* __restrict__ out_q,        // [32768][128]
    float* __restrict__ out_idx,      // [32768] (as float)
    float* __restrict__ out_loss)     // [1]
{
    // tileB[buf][s][lane] = {cb[4s+koff][n], cb[4s+koff+1][n]}  (ISA B layout)
    __shared__ v2f tileB[2][32 * 32];         // 16 KB
    __shared__ int sIdx[WAVES_PER_WG][32];

    const int tid  = threadIdx.x;
    const int wave = tid >> 5;                // 0..3
    const int lane = tid & 31;
    const int l16  = lane & 15;
    const int koff = (lane >> 4) << 1;        // 0 for lanes 0-15, 2 for 16-31
    const int tokBase = blockIdx.x * TOK_PER_WG + wave * TOK_PER_WAVE;

    // --- Preload two A tiles (32 tokens x 128 dims): 64 aligned b64 loads.
    v2f A0[32], A1[32];
    {
        const float* xr0 = x + (size_t)(tokBase + l16) * D_DIM + koff;
        const float* xr1 = xr0 + 16 * D_DIM;
        #pragma unroll
        for (int s = 0; s < 32; ++s) {
            A0[s] = *(const v2f*)(xr0 + 4 * s);
            A1[s] = *(const v2f*)(xr1 + 4 * s);
        }
    }

    // This thread stages steps s = wave + 4j (j=0..7) at its own lane slot.
    const float* gbase = cb + (size_t)(4 * wave + koff) * K_CB + l16;

    float runMinA[8], runMinB[8];
    int   runIdxA[8], runIdxB[8];
    #pragma unroll
    for (int r = 0; r < 8; ++r) {
        runMinA[r] = 3.402823e38f; runIdxA[r] = 0;
        runMinB[r] = 3.402823e38f; runIdxB[r] = 0;
    }

    // --- Pre-stage tile 0 into buffer 0.
    v2f st[8];
    #pragma unroll
    for (int j = 0; j < 8; ++j) {
        const float* p = gbase + (size_t)(16 * j) * K_CB;   // row 4*(wave+4j)+koff
        st[j].x = p[0]; st[j].y = p[K_CB];
    }
    #pragma unroll
    for (int j = 0; j < 8; ++j)
        tileB[0][(wave + 4 * j) * 32 + lane] = st[j];

    for (int nt = 0; nt < NTILES; ++nt) {
        const int cur = nt & 1;
        __syncthreads();                       // tile[cur] fully staged

        // --- Issue global loads for tile nt+1 (latency hidden by WMMAs).
        if (nt + 1 < NTILES) {
            const float* g = gbase + (size_t)(nt + 1) * 16;
            #pragma unroll
            for (int j = 0; j < 8; ++j) {
                const float* p = g + (size_t)(16 * j) * K_CB;
                st[j].x = p[0]; st[j].y = p[K_CB];
            }
        }

        // --- Two 16x16 score tiles: 64 WMMAs; each B pair feeds 4 WMMAs.
        const v2f* tb = &tileB[cur][lane];
        v8f ca = {}, cbc = {};
        #pragma unroll
        for (int s = 0; s < 32; s += 2) {
            v2f B0 = tb[s * 32];
            v2f B1 = tb[(s + 1) * 32];
            ca  = WMMA_F32(A0[s],     B0, ca);
            cbc = WMMA_F32(A1[s],     B0, cbc);
            ca  = WMMA_F32(A0[s + 1], B1, ca);
            cbc = WMMA_F32(A1[s + 1], B1, cbc);
        }

        // --- Store staged data into the other buffer (loads have landed).
        if (nt + 1 < NTILES) {
            #pragma unroll
            for (int j = 0; j < 8; ++j)
                tileB[cur ^ 1][(wave + 4 * j) * 32 + lane] = st[j];
        }

        // --- dist = cb_sq[n] - 2*dot ; running argmin per accumulator row.
        float cq  = cbsq[nt * 16 + l16];
        int   idx = nt * 16 + l16;
        #pragma unroll
        for (int r = 0; r < 8; ++r) {
            float da = fmaf(-2.0f, ca[r],  cq);
            float db = fmaf(-2.0f, cbc[r], cq);
            if (da < runMinA[r]) { runMinA[r] = da; runIdxA[r] = idx; }
            if (db < runMinB[r]) { runMinB[r] = db; runIdxB[r] = idx; }
        }
    }

    // --- Cross-lane argmin within each 16-lane half (token groups),
    //     first-index tie-break to match jnp.argmin.
    #pragma unroll
    for (int m = 1; m < 16; m <<= 1) {
        #pragma unroll
        for (int r = 0; r < 8; ++r) {
            float oMin = __shfl_xor(runMinA[r], m, 32);
            int   oIdx = __shfl_xor(runIdxA[r], m, 32);
            if (oMin < runMinA[r] || (oMin == runMinA[r] && oIdx < runIdxA[r])) {
                runMinA[r] = oMin; runIdxA[r] = oIdx;
            }
            oMin = __shfl_xor(runMinB[r], m, 32);
            oIdx = __shfl_xor(runIdxB[r], m, 32);
            if (oMin < runMinB[r] || (oMin == runMinB[r] && oIdx < runIdxB[r])) {
                runMinB[r] = oMin; runIdxB[r] = oIdx;
            }
        }
    }
    if (l16 == 0) {
        int hi = lane >> 4;                    // 0: tokens r, 1: tokens r+8
        #pragma unroll
        for (int r = 0; r < 8; ++r) {
            sIdx[wave][r + 8 * hi]      = runIdxA[r];   // A tile: tokens 0..15
            sIdx[wave][16 + r + 8 * hi] = runIdxB[r];   // B tile: tokens 16..31
        }
    }
    __syncthreads();

    // --- Dequantize + loss. Each wave handles its own 32 tokens.
    //     Lane covers d = 4*lane..4*lane+3 -> b128 loads/stores.
    float lossAcc = 0.0f;
    #pragma unroll 1
    for (int t = 0; t < TOK_PER_WAVE; ++t) {
        int idx   = sIdx[wave][t];
        int token = tokBase + t;
        const v4f* crow = (const v4f*)(cbT + (size_t)idx * D_DIM);
        const v4f* xrow = (const v4f*)(x   + (size_t)token * D_DIM);
        v4f*       orow = (v4f*)(out_q + (size_t)token * D_DIM);
        v4f cv = crow[lane];
        v4f xv = xrow[lane];
        orow[lane] = cv;                       // straight-through value
        #pragma unroll
        for (int j = 0; j < 4; ++j) {
            float df = cv[j] - xv[j];
            lossAcc  = fmaf(df, df, lossAcc);
        }
        if (lane == 0) out_idx[token] = (float)idx;
    }
    #pragma unroll
    for (int m = 1; m < 32; m <<= 1) lossAcc += __shfl_xor(lossAcc, m, 32);
    if (lane == 0) atomicAdd(out_loss, lossAcc * (1.0f / (float)TOTAL_ELEMS));
}

// ---------------------------------------------------------------------------
extern "C" void kernel_launch(void* const* d_in, const int* in_sizes, int n_in,
                              void* d_out, int out_size, void* d_ws, size_t ws_size,
                              hipStream_t stream)
{
    const float* x  = (const float*)d_in[0];   // [8,4096,128]
    const float* cb = (const float*)d_in[1];   // [128,1024]

    float* out      = (float*)d_out;
    float* out_q    = out;                         // 4194304 floats
    float* out_idx  = out + (size_t)TOTAL_ELEMS;   // 32768 floats
    float* out_loss = out_idx + N_TOKENS;          // 1 float

    float* cbT  = (float*)d_ws;                    // 1024*128 floats = 512 KB
    float* cbsq = cbT + (size_t)K_CB * D_DIM;      // 1024 floats

    vq_prep_kernel<<<K_CB / 256, 256, 0, stream>>>(cb, cbT, cbsq, out_loss);
    vq_main_kernel<<<N_TOKENS / TOK_PER_WG, THREADS_PER_WG, 0, stream>>>(
        x, cb, cbT, cbsq, out_q, out_idx, out_loss);
}